// ConditionalStructureEncoder_44573170598274
// MI455X (gfx1250) — compile-verified
//
#include <hip/hip_runtime.h>
#include <hip/hip_bf16.h>

typedef __attribute__((ext_vector_type(2))) float v2f;
typedef __attribute__((ext_vector_type(8))) float v8f;

#define CDIV(a, b) (((a) + (b) - 1) / (b))

__device__ __forceinline__ void fadd_atomic(float* p, float v) {
#if defined(__AMDGCN__)
    unsafeAtomicAdd(p, v);   // guaranteed global_atomic_add_f32
#else
    atomicAdd(p, v);
#endif
}

// ---------------------------------------------------------------------------
// Tiny conditioning MLP: hom = relu(hc @ hw1 + hb1) @ hw2 + hb2   (1x128)
// ---------------------------------------------------------------------------
__global__ void hom_kernel(const float* __restrict__ hc,
                           const float* __restrict__ hw1, const float* __restrict__ hb1,
                           const float* __restrict__ hw2, const float* __restrict__ hb2,
                           float* __restrict__ hom) {
    __shared__ float t[64];
    int j = threadIdx.x;
    if (j < 64) {
        float s = hb1[j];
        #pragma unroll
        for (int i = 0; i < 3; ++i) s += hc[i] * hw1[i * 64 + j];
        t[j] = fmaxf(s, 0.0f);
    }
    __syncthreads();
    float s = hb2[j];
    #pragma unroll 8
    for (int k = 0; k < 64; ++k) s += t[k] * hw2[k * 128 + j];
    hom[j] = s;
}

// ---------------------------------------------------------------------------
// Degree / norm
// ---------------------------------------------------------------------------
__global__ void deg_init_kernel(float* __restrict__ deg, int n) {
    int i = blockIdx.x * blockDim.x + threadIdx.x;
    if (i < n) deg[i] = 1.0f;   // self loop contributes 1
}

__global__ void deg_accum_kernel(const int* __restrict__ ei, int E, float* __restrict__ deg) {
    int e = blockIdx.x * blockDim.x + threadIdx.x;
    if (e < E) fadd_atomic(&deg[ei[E + e]], 1.0f);   // dst row
}

__global__ void deg_rsqrt_kernel(float* __restrict__ dis, int n) {
    int i = blockIdx.x * blockDim.x + threadIdx.x;
    if (i < n) dis[i] = rsqrtf(dis[i]);              // deg >= 1 always
}

// ---------------------------------------------------------------------------
// Zero an N*128 float buffer (float4 grid-stride).
// ---------------------------------------------------------------------------
__global__ void zero_kernel(float4* __restrict__ p, long n4) {
    long i = (long)blockIdx.x * blockDim.x + threadIdx.x;
    long stride = (long)gridDim.x * blockDim.x;
    float4 z = make_float4(0.f, 0.f, 0.f, 0.f);
    for (; i < n4; i += stride) p[i] = z;
}

// ---------------------------------------------------------------------------
// WMMA fp32 GEMM:  C[nrows x OUT_W] = (A [+ addv broadcast]) @ W  [+ bias]
// A: nrows x 128 row-major, W: 128 x OUT_W row-major.
// One wave -> one 16x16 tile via V_WMMA_F32_16X16X4_F32 (K stepped by 4).
// blockDim.x = 32 * (OUT_W/16); blockIdx.x = 16-row strip.
// ---------------------------------------------------------------------------
template <int OUT_W, bool ADD_VEC, bool HAS_BIAS>
__global__ __launch_bounds__(32 * (OUT_W / 16)) void
gemm_wmma_kernel(const float* __restrict__ A, const float* __restrict__ W,
                 const float* __restrict__ addv, const float* __restrict__ bias,
                 float* __restrict__ C, int nrows) {
    const int wave = threadIdx.x >> 5;
    const int lane = threadIdx.x & 31;
    const int m0 = blockIdx.x * 16;
    if (m0 >= nrows) return;
    const int n0 = wave * 16;
    const int half = lane >> 4;          // 0: K+{0,1}, 1: K+{2,3}
    const int col = n0 + (lane & 15);
    int row = m0 + (lane & 15);
    if (row >= nrows) row = nrows - 1;   // clamp loads (partial tile only)

    const float* __restrict__ arow = A + (size_t)row * 128;
    v8f acc = {};
    #pragma unroll 8
    for (int k = 0; k < 128; k += 4) {
        const int ka = k + 2 * half;
        v2f a, b;
        float a0 = arow[ka];
        float a1 = arow[ka + 1];
        if (ADD_VEC) { a0 += addv[ka]; a1 += addv[ka + 1]; }
        a.x = a0; a.y = a1;
        b.x = W[(size_t)ka * OUT_W + col];
        b.y = W[(size_t)(ka + 1) * OUT_W + col];
        acc = __builtin_amdgcn_wmma_f32_16x16x4_f32(
            /*neg_a=*/false, a, /*neg_b=*/false, b,
            /*c_mod=*/(short)0, acc, /*reuse_a=*/false, /*reuse_b=*/false);
    }
    const float bv = HAS_BIAS ? bias[col] : 0.0f;
    float* __restrict__ cbase = C + (size_t)(m0 + 8 * half) * OUT_W + col;
    if (m0 + 16 <= nrows) {
        // full tile: unguarded stores (only path when nrows % 16 == 0)
        #pragma unroll
        for (int v = 0; v < 8; ++v) {
            float outv = acc[v];
            if (HAS_BIAS) outv += bv;    // compile-time: no +0 add otherwise
            cbase[(size_t)v * OUT_W] = outv;
        }
    } else {
        #pragma unroll
        for (int v = 0; v < 8; ++v) {
            const int r = m0 + v + 8 * half;
            if (r < nrows) {
                float outv = acc[v];
                if (HAS_BIAS) outv += bv;
                C[(size_t)r * OUT_W + col] = outv;
            }
        }
    }
}

// ---------------------------------------------------------------------------
// Edge scatter: agg[dst] += hW[src] * dis[src]*dis[dst]
// One wave per edge (edge data is wave-uniform -> readfirstlane so dis[]
// loads scalarize); float4 gather + 4 hw fp32 atomics per lane.
// ---------------------------------------------------------------------------
__global__ void scatter_kernel(const float* __restrict__ hW, float* __restrict__ agg,
                               const int* __restrict__ ei, const float* __restrict__ dis,
                               int E) {
    long tid = (long)blockIdx.x * blockDim.x + threadIdx.x;
    long e = tid >> 5;
    if (e >= E) return;
    const int lane = threadIdx.x & 31;
    const int s = __builtin_amdgcn_readfirstlane(ei[e]);
    const int d = __builtin_amdgcn_readfirstlane(ei[(long)E + e]);
    const float w = dis[s] * dis[d];
    const float4 v = *((const float4*)(hW + (size_t)s * 128) + lane);
    float* o = agg + (size_t)d * 128 + (size_t)lane * 4;
    fadd_atomic(o + 0, v.x * w);
    fadd_atomic(o + 1, v.y * w);
    fadd_atomic(o + 2, v.z * w);
    fadd_atomic(o + 3, v.w * w);
}

// ---------------------------------------------------------------------------
// Finish a GCN layer: out = relu(agg + hW * dis^2 (self loop) + bias)
// float4 over N*32 quads.
// ---------------------------------------------------------------------------
__global__ void finish_layer_kernel(const float4* __restrict__ agg, const float4* __restrict__ hW,
                                    const float* __restrict__ dis, const float4* __restrict__ bias4,
                                    float4* __restrict__ out, long total4) {
    long i = (long)blockIdx.x * blockDim.x + threadIdx.x;
    long stride = (long)gridDim.x * blockDim.x;
    for (; i < total4; i += stride) {
        const int node = (int)(i >> 5);      // 32 quads per node
        const int fq = (int)(i & 31);
        const float dv = dis[node];
        const float d2 = dv * dv;
        const float4 a = agg[i];
        const float4 h = hW[i];
        const float4 b = bias4[fq];
        float4 r;
        r.x = fmaxf(a.x + h.x * d2 + b.x, 0.0f);
        r.y = fmaxf(a.y + h.y * d2 + b.y, 0.0f);
        r.z = fmaxf(a.z + h.z * d2 + b.z, 0.0f);
        r.w = fmaxf(a.w + h.w * d2 + b.w, 0.0f);
        out[i] = r;
    }
}

// ---------------------------------------------------------------------------
extern "C" void kernel_launch(void* const* d_in, const int* in_sizes, int n_in,
                              void* d_out, int out_size, void* d_ws, size_t ws_size,
                              hipStream_t stream) {
    const float* x    = (const float*)d_in[0];
    const int*   ei   = (const int*)d_in[1];
    const float* hc   = (const float*)d_in[2];
    const float* hw1  = (const float*)d_in[3];
    const float* hb1  = (const float*)d_in[4];
    const float* hw2  = (const float*)d_in[5];
    const float* hb2  = (const float*)d_in[6];
    const float* gw1  = (const float*)d_in[7];
    const float* gb1  = (const float*)d_in[8];
    const float* gw2  = (const float*)d_in[9];
    const float* gb2  = (const float*)d_in[10];
    const float* muw  = (const float*)d_in[11];
    const float* mub  = (const float*)d_in[12];
    const float* lvw  = (const float*)d_in[13];
    const float* lvb  = (const float*)d_in[14];

    const int N = in_sizes[0] / 128;
    const int E = in_sizes[1] / 2;
    const long NF = (long)N * 128;

    // Workspace layout (floats)
    float* ws   = (float*)d_ws;
    float* hom  = ws;                 // 128
    float* dis  = hom + 128;          // N  (deg, then rsqrt in place)
    float* bufA = dis + N;            // N*128: GEMM output hW
    float* bufB = bufA + NF;          // N*128: aggregation target
    float* bufH = bufB + NF;          // N*128: activated features

    float* out_mu = (float*)d_out;
    float* out_lv = out_mu + (long)N * 64;

    const int rowTiles = CDIV(N, 16);

    // 0) conditioning vector
    hom_kernel<<<1, 128, 0, stream>>>(hc, hw1, hb1, hw2, hb2, hom);

    // 1) degrees -> dis
    deg_init_kernel<<<CDIV(N, 256), 256, 0, stream>>>(dis, N);
    deg_accum_kernel<<<CDIV(E, 256), 256, 0, stream>>>(ei, E, dis);
    deg_rsqrt_kernel<<<CDIV(N, 256), 256, 0, stream>>>(dis, N);

    // ---- GCN layer 1: hW = (x + hom) @ gw1 ----
    gemm_wmma_kernel<128, true, false><<<rowTiles, 256, 0, stream>>>(x, gw1, hom, nullptr, bufA, N);
    zero_kernel<<<2048, 256, 0, stream>>>((float4*)bufB, NF / 4);
    scatter_kernel<<<CDIV((long)E * 32, 256), 256, 0, stream>>>(bufA, bufB, ei, dis, E);
    finish_layer_kernel<<<4096, 256, 0, stream>>>((const float4*)bufB, (const float4*)bufA,
                                                  dis, (const float4*)gb1, (float4*)bufH, NF / 4);

    // ---- GCN layer 2: hW = h1 @ gw2 ----
    gemm_wmma_kernel<128, false, false><<<rowTiles, 256, 0, stream>>>(bufH, gw2, nullptr, nullptr, bufA, N);
    zero_kernel<<<2048, 256, 0, stream>>>((float4*)bufB, NF / 4);
    scatter_kernel<<<CDIV((long)E * 32, 256), 256, 0, stream>>>(bufA, bufB, ei, dis, E);
    finish_layer_kernel<<<4096, 256, 0, stream>>>((const float4*)bufB, (const float4*)bufA,
                                                  dis, (const float4*)gb2, (float4*)bufH, NF / 4);

    // ---- heads ----
    gemm_wmma_kernel<64, false, true><<<rowTiles, 128, 0, stream>>>(bufH, muw, nullptr, mub, out_mu, N);
    gemm_wmma_kernel<64, false, true><<<rowTiles, 128, 0, stream>>>(bufH, lvw, nullptr, lvb, out_lv, N);
}